// ThreatGNN_82325933130190
// MI455X (gfx1250) — compile-verified
//
#include <hip/hip_runtime.h>

#define DH     64
#define NGRAPH 64
#define DOUT   8
#define EPSBN  1e-5f

typedef float v2f __attribute__((ext_vector_type(2)));
typedef float v4f __attribute__((ext_vector_type(4)));
typedef float v8f __attribute__((ext_vector_type(8)));

// ---------------------------------------------------------------- utilities
__global__ void gnn_zero_kernel(float* __restrict__ p, long n) {
    long i = (long)blockIdx.x * blockDim.x + threadIdx.x;
    long stride = (long)gridDim.x * blockDim.x;
    for (; i < n; i += stride) p[i] = 0.0f;
}

// deg[dst] += 1 for every edge (float is exact for these counts)
__global__ void gnn_degree_kernel(const int* __restrict__ dst, float* __restrict__ deg, int E) {
    int e = blockIdx.x * blockDim.x + threadIdx.x;
    if (e < E) atomicAdd(&deg[dst[e]], 1.0f);
}

// in-place: deg -> rsqrt(deg + 1)   (self-loop normalization)
__global__ void gnn_dinv_kernel(float* __restrict__ deg, int N) {
    int i = blockIdx.x * blockDim.x + threadIdx.x;
    if (i < N) deg[i] = rsqrtf(deg[i] + 1.0f);
}

// ---------------------------------------------------------------- H = X @ W
// X: [N,64] f32 row-major, W: [64,64] f32, H: [N,64] f32.
// One wave32 per 16-row tile; 4 f32 16x16 accumulators cover N=64;
// K walked in steps of 4 via V_WMMA_F32_16X16X4_F32 (full fp32 precision).
// W is pre-swizzled into LDS in per-fragment order so every B fragment is a
// single aligned ds_load_b64 (32 lanes x 8B contiguous -> bank-conflict-free).
__global__ __launch_bounds__(256) void gnn_gemm64_wmma(
        const float* __restrict__ X, const float* __restrict__ W,
        float* __restrict__ H, int N) {
    __shared__ float wsh[DH * DH];   // 64 fragments * 32 lanes * 2 floats = 16 KB
    // fragment f = (kk/4)*4 + nt ; lane pair = {W[kb][col], W[kb+1][col]}
    // with kb = kk + 2*(lane>=16), col = nt*16 + (lane&15)
    for (int i = threadIdx.x; i < (DH * DH) / 2; i += blockDim.x) {
        int frag = i >> 5;
        int ln   = i & 31;
        int kk   = (frag >> 2) << 2;
        int nt   = frag & 3;
        int kb   = kk + ((ln >> 4) << 1);
        int col  = nt * 16 + (ln & 15);
        v2f w;
        w.x = W[kb * DH + col];
        w.y = W[(kb + 1) * DH + col];
        *reinterpret_cast<v2f*>(&wsh[i * 2]) = w;
    }
    __syncthreads();

    const int lane = threadIdx.x & 31;
    const int wid  = threadIdx.x >> 5;
    const int m0   = (blockIdx.x * 8 + wid) * 16;   // tile base, uniform per wave
    if (m0 >= N) return;

    const int mrow  = m0 + (lane & 15);   // A: both lane halves cover M=0..15
    const int khalf = (lane >> 4) * 2;    // A: K pairs split across lane halves
    const int ncol  = lane & 15;

    v8f acc[4];
#pragma unroll
    for (int t = 0; t < 4; ++t)
#pragma unroll
        for (int r = 0; r < 8; ++r) acc[t][r] = 0.0f;

    const float* xrow = X + (long)mrow * DH;
#pragma unroll
    for (int k4 = 0; k4 < DH / 4; ++k4) {
        v2f a = *reinterpret_cast<const v2f*>(xrow + k4 * 4 + khalf);  // 8B aligned
#pragma unroll
        for (int nt = 0; nt < 4; ++nt) {
            v2f b = *reinterpret_cast<const v2f*>(
                &wsh[((k4 * 4 + nt) * 32 + lane) * 2]);                // ds_load_b64
            acc[nt] = __builtin_amdgcn_wmma_f32_16x16x4_f32(
                false, a, false, b, (short)0, acc[nt], false, false);
        }
    }

    // C/D layout: VGPR r -> M = r + 8*(lane>=16), N = lane&15
    const int rbase = (lane >> 4) * 8;
#pragma unroll
    for (int nt = 0; nt < 4; ++nt)
#pragma unroll
        for (int r = 0; r < 8; ++r)
            H[(long)(m0 + rbase + r) * DH + nt * 16 + ncol] = acc[nt][r];
}

// agg = h*dinv^2 + bias  (seeds self-loop + bias before the edge atomics)
__global__ void gnn_init_agg_kernel(
        float* __restrict__ agg, const float* __restrict__ h,
        const float* __restrict__ dinv, const float* __restrict__ bias, int N) {
    long idx = (long)blockIdx.x * blockDim.x + threadIdx.x;
    if (idx >= (long)N * DH) return;
    int i = (int)(idx >> 6);
    int c = (int)(idx & 63);
    float di = dinv[i];
    agg[idx] = h[idx] * di * di + bias[c];
}

// ------------------------------------------------- edge scatter (dominant cost)
// 64 threads per edge: coalesced 256B gather of h[src,:], coalesced f32 atomics
// into agg[dst,:].
__global__ void gnn_edge_agg_kernel(
        const float* __restrict__ h, const float* __restrict__ dinv,
        const int* __restrict__ src, const int* __restrict__ dst,
        float* __restrict__ agg, int E) {
    long idx = (long)blockIdx.x * blockDim.x + threadIdx.x;
    long e = idx >> 6;
    int  c = (int)(idx & 63);
    if (e >= E) return;
    int s = src[e];
    int d = dst[e];
    float norm = dinv[s] * dinv[d];
    atomicAdd(&agg[(long)d * DH + c], h[(long)s * DH + c] * norm);
}

// fused: x = relu(x) in place, plus per-column sum/sumsq for BN
// (block-level LDS reduction -> 2 atomics per column per block)
__global__ __launch_bounds__(256) void gnn_relu_stats_kernel(
        float* __restrict__ x, float* __restrict__ stats, int N) {
    __shared__ float s1[4][DH];
    __shared__ float s2[4][DH];
    int c  = threadIdx.x & 63;
    int yr = threadIdx.x >> 6;                      // 0..3
    float a = 0.0f, b = 0.0f;
    for (long i = (long)blockIdx.x * 4 + yr; i < N; i += (long)gridDim.x * 4) {
        float v = fmaxf(x[i * DH + c], 0.0f);
        x[i * DH + c] = v;
        a += v;
        b += v * v;
    }
    s1[yr][c] = a;
    s2[yr][c] = b;
    __syncthreads();
    if (yr == 0) {
        a = s1[0][c] + s1[1][c] + s1[2][c] + s1[3][c];
        b = s2[0][c] + s2[1][c] + s2[2][c] + s2[3][c];
        atomicAdd(&stats[c], a);
        atomicAdd(&stats[DH + c], b);
    }
}

__global__ void gnn_bn_apply_kernel(
        float* __restrict__ x, const float* __restrict__ stats,
        const float* __restrict__ gamma, const float* __restrict__ beta, int N) {
    long idx = (long)blockIdx.x * blockDim.x + threadIdx.x;
    if (idx >= (long)N * DH) return;
    int c = (int)(idx & 63);
    float invN = 1.0f / (float)N;
    float mu   = stats[c] * invN;
    float var  = stats[DH + c] * invN - mu * mu;    // biased variance, matches reference
    x[idx] = (x[idx] - mu) * rsqrtf(var + EPSBN) * gamma[c] + beta[c];
}

// segment-sum pooling with ReLU applied inline (layer-3 activation); no
// write-back of h3 needed since only the pooled result is consumed.
__global__ void gnn_pool_kernel(
        const float* __restrict__ h, const int* __restrict__ batch,
        float* __restrict__ psum, float* __restrict__ pcnt, int N) {
    long idx = (long)blockIdx.x * blockDim.x + threadIdx.x;
    if (idx >= (long)N * DH) return;
    int i = (int)(idx >> 6);
    int c = (int)(idx & 63);
    int g = batch[i];
    atomicAdd(&psum[(long)g * DH + c], fmaxf(h[idx], 0.0f));
    if (c == 0) atomicAdd(&pcnt[g], 1.0f);
}

// out[g,o] = (psum[g,:]/max(cnt,1)) @ Wc + bc   -- 512 outputs, one block
__global__ void gnn_classifier_kernel(
        const float* __restrict__ psum, const float* __restrict__ pcnt,
        const float* __restrict__ Wc, const float* __restrict__ bc,
        float* __restrict__ out) {
    int t = threadIdx.x;                            // 0..511
    int g = t >> 3;
    int o = t & 7;
    float inv = 1.0f / fmaxf(pcnt[g], 1.0f);
    float acc = bc[o];
#pragma unroll
    for (int d = 0; d < DH; ++d)
        acc += psum[g * DH + d] * inv * Wc[d * DOUT + o];
    out[g * DOUT + o] = acc;
}

// ---------------------------------------------------------------- launcher
extern "C" void kernel_launch(void* const* d_in, const int* in_sizes, int n_in,
                              void* d_out, int out_size, void* d_ws, size_t ws_size,
                              hipStream_t stream) {
    const float* x     = (const float*)d_in[0];
    const int*   ei    = (const int*)  d_in[1];
    const int*   batch = (const int*)  d_in[2];
    const float* W1    = (const float*)d_in[3];
    const float* b1    = (const float*)d_in[4];
    const float* W2    = (const float*)d_in[5];
    const float* b2    = (const float*)d_in[6];
    const float* W3    = (const float*)d_in[7];
    const float* b3    = (const float*)d_in[8];
    const float* gamma = (const float*)d_in[9];
    const float* beta  = (const float*)d_in[10];
    const float* Wc    = (const float*)d_in[11];
    const float* bc    = (const float*)d_in[12];
    float* out = (float*)d_out;

    const int N = in_sizes[0] / DH;
    const int E = in_sizes[1] / 2;
    const int* src = ei;
    const int* dst = ei + E;

    // workspace layout (all 256B-aligned)
    char* ws = (char*)d_ws;
    auto take = [&](size_t bytes) {
        char* p = ws;
        ws += (bytes + 255) & ~(size_t)255;
        return p;
    };
    float* dinv  = (float*)take((size_t)N * 4);
    float* B0    = (float*)take((size_t)N * DH * 4);   // h = in @ W
    float* B1    = (float*)take((size_t)N * DH * 4);   // aggregation / layer output
    float* stats = (float*)take(2 * DH * 4);
    float* psum  = (float*)take((size_t)NGRAPH * DH * 4);
    float* pcnt  = (float*)take((size_t)NGRAPH * 4);

    const long NF      = (long)N * DH;
    const int  blkNF   = (int)((NF + 255) / 256);
    const int  blkN    = (N + 255) / 256;
    const int  blkE    = (E + 255) / 256;
    const int  blkEdge = (int)(((long)E * DH + 255) / 256);
    const int  blkGemm = ((N / 16) + 7) / 8;

    // degree -> dinv
    gnn_zero_kernel<<<blkN, 256, 0, stream>>>(dinv, N);
    gnn_degree_kernel<<<blkE, 256, 0, stream>>>(dst, dinv, E);
    gnn_dinv_kernel<<<blkN, 256, 0, stream>>>(dinv, N);

    const float* lin[3] = {x, B1, B1};
    const float* lW[3]  = {W1, W2, W3};
    const float* lb[3]  = {b1, b2, b3};

    for (int l = 0; l < 3; ++l) {
        gnn_gemm64_wmma<<<blkGemm, 256, 0, stream>>>(lin[l], lW[l], B0, N);
        // seed agg with self-loop + bias (input buffer is dead after the GEMM)
        gnn_init_agg_kernel<<<blkNF, 256, 0, stream>>>(B1, B0, dinv, lb[l], N);
        gnn_edge_agg_kernel<<<blkEdge, 256, 0, stream>>>(B0, dinv, src, dst, B1, E);
        if (l < 2) {
            gnn_zero_kernel<<<1, 256, 0, stream>>>(stats, 2 * DH);
            gnn_relu_stats_kernel<<<512, 256, 0, stream>>>(B1, stats, N);
            gnn_bn_apply_kernel<<<blkNF, 256, 0, stream>>>(B1, stats, gamma, beta, N);
        }
    }

    // pooling (ReLU fused) + classifier
    gnn_zero_kernel<<<1, 256, 0, stream>>>(psum, NGRAPH * DH);
    gnn_zero_kernel<<<1, 256, 0, stream>>>(pcnt, NGRAPH);
    gnn_pool_kernel<<<blkNF, 256, 0, stream>>>(B1, batch, psum, pcnt, N);
    gnn_classifier_kernel<<<1, NGRAPH * DOUT, 0, stream>>>(psum, pcnt, Wc, bc, out);
}